// FFEdgeCountingLayer_19593640804419
// MI455X (gfx1250) — compile-verified
//
#include <hip/hip_runtime.h>
#include <stdint.h>

// Problem shape (fixed by setup_inputs)
#define B_DIM 256
#define I_DIM 512
#define O_DIM 512

// Tiling
#define O_TILE 16
#define B_TILE 32
// LDS row stride for coefficient tile, in float2 units.
// 514*8 = 4112 bytes: 16B aligned and 4-bank skew per row -> 16 o-rows land
// on 16 distinct bank pairs (conflict-free b64 reads).
#define AC_STRIDE 514

typedef uint32_t u32x4 __attribute__((ext_vector_type(4)));
typedef uint32_t u32x8 __attribute__((ext_vector_type(8)));

// ---------------- CDNA5 async global->LDS copy (ASYNCcnt path) ---------------
__device__ __forceinline__ void async_load_b128(uint32_t lds_addr, const void* gaddr) {
  // VGLOBAL async load: VDST holds the per-lane LDS byte address, VADDR the 64-bit global addr.
  asm volatile("global_load_async_to_lds_b128 %0, %1, off"
               :: "v"(lds_addr), "v"((uint64_t)(uintptr_t)gaddr)
               : "memory");
}

__device__ __forceinline__ void wait_async0() {
#if __has_builtin(__builtin_amdgcn_s_wait_asynccnt)
  __builtin_amdgcn_s_wait_asynccnt(0);
#else
  asm volatile("s_wait_asynccnt 0x0" ::: "memory");
#endif
}

// ---------------- CDNA5 Tensor Data Mover (TENSORcnt path) -------------------
// 1D contiguous DMA: n_elems f32 from gaddr to the LDS address in lds_addr. Issued once per
// wave (EXEC ignored by TDM); all fields wave-uniform -> SGPR descriptor.
__device__ __forceinline__ void tdm_load_1d_f32(uint32_t lds_addr, const void* gaddr,
                                                uint32_t n_elems) {
  uint64_t ga = (uint64_t)(uintptr_t)gaddr;
  u32x4 g0;
  g0[0] = 1u;                                                  // count=1 (valid), user mode
  g0[1] = lds_addr;                                            // lds_addr[31:0]
  g0[2] = (uint32_t)ga;                                        // global_addr[31:0]
  g0[3] = (uint32_t)((ga >> 32) & 0x01FFFFFFu) | 0x80000000u;  // global_addr[56:32] | type=2
  u32x8 g1;
  g1[0] = 2u << 16;                                  // wg_mask=0, data_size=2 (4 bytes)
  g1[1] = (n_elems & 0xFFFFu) << 16;                 // tensor_dim0[15:0]
  g1[2] = (n_elems >> 16) | (1u << 16);              // tensor_dim0[31:16] | tensor_dim1=1
  g1[3] = (n_elems & 0xFFFFu) << 16;                 // tile_dim0 (16-bit)
  g1[4] = 1u;                                        // tile_dim1=1, tile_dim2=0
  g1[5] = n_elems;                                   // tensor_dim0_stride[31:0]
  g1[6] = 0u;                                        // stride hi / dim1_stride lo
  g1[7] = 0u;
  // 2-group form: VADDR2/VADDR3 omitted (NULL) -> tensor up to 2D.
  asm volatile("tensor_load_to_lds %0, %1" :: "s"(g0), "s"(g1) : "memory");
}

__device__ __forceinline__ void wait_tensor0() {
#if __has_builtin(__builtin_amdgcn_s_wait_tensorcnt)
  __builtin_amdgcn_s_wait_tensorcnt(0);
#else
  asm volatile("s_wait_tensorcnt 0x0" ::: "memory");
#endif
}

// ---------------- Phase 1: hard gumbel-softmax -> per-edge affine (a, c) -----
__device__ __forceinline__ float gumbel_noise(uint32_t idx) {
  uint32_t v = idx * 0x9E3779B9u + 42u * 0x85EBCA6Bu;
  v ^= v >> 16; v *= 0x7FEB352Du;
  v ^= v >> 15; v *= 0x846CA68Bu;
  v ^= v >> 16;
  float u = ((float)(v >> 8) + 0.5f) * (1.0f / 16777216.0f);   // open (0,1)
  return -__logf(-__logf(u));
}

__global__ void __launch_bounds__(256)
sel_to_affine_kernel(const float* __restrict__ etc, float2* __restrict__ ac, int n_edges) {
  int e = blockIdx.x * 256 + threadIdx.x;
  if (e >= n_edges) return;
  int base = e * 3;
  float v0 = etc[base + 0] + gumbel_noise((uint32_t)base + 0u);
  float v1 = etc[base + 1] + gumbel_noise((uint32_t)base + 1u);
  float v2 = etc[base + 2] + gumbel_noise((uint32_t)base + 2u);
  float2 r;
  if (v0 >= v1 && v0 >= v2)      r = make_float2( 1.0f, 0.0f);  // identity:  x
  else if (v1 >= v2)             r = make_float2(-1.0f, 1.0f);  // negation:  1-x
  else                           r = make_float2( 0.0f, 1.0f);  // no_edge:   1
  ac[e] = r;
}

// ---------------- Phase 2: out[b,o] = min_i fma(a[o,i], x[b,i], c[o,i]) -----
__global__ void __launch_bounds__(256)
fuzzy_min_kernel(const float* __restrict__ x, const float2* __restrict__ ac,
                 float* __restrict__ out) {
  extern __shared__ char smem[];
  float*  x_s  = (float*)smem;                          // [B_TILE][I_DIM]   64 KB
  float2* ac_s = (float2*)(smem + B_TILE * I_DIM * 4);  // [O_TILE][AC_STRIDE] ~64 KB

  const int tid    = threadIdx.x;
  const int o_base = blockIdx.x * O_TILE;
  const int b_base = blockIdx.y * B_TILE;

  // --- x tile via TDM: 8 waves each DMA one contiguous 8 KB slice (the whole
  //     tile is one contiguous 64 KB global region: full i-range, 32 b rows).
  {
    const int wave = tid >> 5;                               // wave-uniform
    uint32_t    lx = (uint32_t)(uintptr_t)x_s + (uint32_t)wave * 8192u;
    const char* gx = (const char*)(x + (size_t)b_base * I_DIM) + wave * 8192;
    tdm_load_1d_f32(lx, gx, 2048u);
  }

  // --- AC tile via per-lane async-to-LDS (4 KB global row == 256 lanes * 16B),
  //     into the bank-skewed padded layout.
  #pragma unroll
  for (int r = 0; r < O_TILE; ++r) {
    const char* ga = (const char*)(ac + (size_t)(o_base + r) * I_DIM);
    uint32_t    la = (uint32_t)(uintptr_t)(ac_s + r * AC_STRIDE);
    async_load_b128(la + (uint32_t)(tid * 16), ga + tid * 16);
  }

  wait_tensor0();
  wait_async0();
  __syncthreads();

  // Thread -> (o_local, two b rows). Lanes 0..15 of a wave share b, differ in o:
  // coefficient LDS reads hit distinct bank pairs, x reads are broadcast.
  const int o_local = tid & (O_TILE - 1);
  const int b_group = tid >> 4;                 // 0..15, two b rows each
  const float*  xr0 = x_s + (2 * b_group + 0) * I_DIM;
  const float*  xr1 = x_s + (2 * b_group + 1) * I_DIM;
  const float2* acr = ac_s + o_local * AC_STRIDE;

  float m0 = 3.0e38f, m1 = 3.0e38f;
  #pragma unroll 4
  for (int i = 0; i < I_DIM; i += 4) {
    float4 xv0 = *(const float4*)(xr0 + i);
    float4 xv1 = *(const float4*)(xr1 + i);
    float2 a0 = acr[i + 0], a1 = acr[i + 1], a2 = acr[i + 2], a3 = acr[i + 3];
    m0 = fminf(m0, fminf(fminf(fmaf(a0.x, xv0.x, a0.y), fmaf(a1.x, xv0.y, a1.y)),
                         fminf(fmaf(a2.x, xv0.z, a2.y), fmaf(a3.x, xv0.w, a3.y))));
    m1 = fminf(m1, fminf(fminf(fmaf(a0.x, xv1.x, a0.y), fmaf(a1.x, xv1.y, a1.y)),
                         fminf(fmaf(a2.x, xv1.z, a2.y), fmaf(a3.x, xv1.w, a3.y))));
  }

  const int o  = o_base + o_local;
  const int b0 = b_base + 2 * b_group;
  out[(size_t)(b0 + 0) * O_DIM + o] = m0;
  out[(size_t)(b0 + 1) * O_DIM + o] = m1;
}

// ---------------------------------- launch ----------------------------------
extern "C" void kernel_launch(void* const* d_in, const int* in_sizes, int n_in,
                              void* d_out, int out_size, void* d_ws, size_t ws_size,
                              hipStream_t stream) {
  (void)in_sizes; (void)n_in; (void)out_size; (void)ws_size;
  const float* x   = (const float*)d_in[0];   // [256, 512]
  const float* etc = (const float*)d_in[1];   // [512, 512, 3]
  float*       out = (float*)d_out;           // [256, 512]
  float2*      ac  = (float2*)d_ws;           // O*I float2 = 2 MB workspace

  const int n_edges = O_DIM * I_DIM;
  sel_to_affine_kernel<<<(n_edges + 255) / 256, 256, 0, stream>>>(etc, ac, n_edges);

  dim3 grid(O_DIM / O_TILE, B_DIM / B_TILE);   // 32 x 8 = 256 workgroups
  size_t smem = (size_t)B_TILE * I_DIM * 4 + (size_t)O_TILE * AC_STRIDE * 8; // 131328 B
  fuzzy_min_kernel<<<grid, 256, smem, stream>>>(x, ac, out);
}